// ContrastiveCenterLoss_70437463654503
// MI455X (gfx1250) — compile-verified
//
#include <hip/hip_runtime.h>

typedef float v2f __attribute__((ext_vector_type(2)));
typedef float v8f __attribute__((ext_vector_type(8)));

#define BATCH 65536
#define DIMH 512
#define NCLS 1000

// ---------------- zero the workspace (hist counts + accumulator) -------------
__global__ void ccl_zero_kernel(unsigned int* __restrict__ ws) {
    int i = threadIdx.x + blockIdx.x * blockDim.x;
    if (i < NCLS + 2) ws[i] = 0u;
}

// ---------------- histogram: counts[c] = #{i : y[i] == c} --------------------
__global__ void ccl_hist_kernel(const int* __restrict__ y,
                                unsigned int* __restrict__ hist) {
    int i = threadIdx.x + blockIdx.x * blockDim.x;
    if (i < BATCH) atomicAdd(&hist[y[i]], 1u);
}

// ---------------- main: per-16-row tile squared distances --------------------
// One wave32 per tile of 16 rows. lane%16 selects the row; lane/16 selects
// which half of the 512 columns this lane walks. Each lane keeps two fp32
// partial sums -> 4 partials per row across the wave, laid out exactly as the
// B operand (4x16) of V_WMMA_F32_16X16X4_F32. A = ones collapses them:
// D[i][j] = sum_k B[k][j] = dist[j]/(count[j]+1) for every i.
__global__ void ccl_main_kernel(const int* __restrict__ y,
                                const float* __restrict__ hidden,
                                const float* __restrict__ centers,
                                const unsigned int* __restrict__ hist,
                                float* __restrict__ accum) {
    const int gtid = blockIdx.x * blockDim.x + threadIdx.x;
    const int wave = gtid >> 5;          // global wave id == tile id
    const int lane = threadIdx.x & 31;
    const int rown = lane & 15;          // row within tile
    const int half = lane >> 4;          // which 256-column half

    const int row = wave * 16 + rown;    // grid sized so row < BATCH always
    const int cls = y[row];
    const float inv = 1.0f / (float)(hist[cls] + 1u);

    const float* __restrict__ hrow = hidden  + (size_t)row * DIMH + half * 256;
    const float* __restrict__ crow = centers + (size_t)cls * DIMH + half * 256;

    float acc0 = 0.0f;
    float acc1 = 0.0f;
#pragma unroll 8
    for (int c = 0; c < 128; c += 4) {            // columns [0,128) of this half
        float4 hv = *(const float4*)(hrow + c);
        float4 cv = *(const float4*)(crow + c);
        float dx = hv.x - cv.x, dy = hv.y - cv.y;
        float dz = hv.z - cv.z, dw = hv.w - cv.w;
        acc0 = fmaf(dx, dx, acc0);
        acc0 = fmaf(dy, dy, acc0);
        acc0 = fmaf(dz, dz, acc0);
        acc0 = fmaf(dw, dw, acc0);
    }
#pragma unroll 8
    for (int c = 128; c < 256; c += 4) {          // columns [128,256) of this half
        float4 hv = *(const float4*)(hrow + c);
        float4 cv = *(const float4*)(crow + c);
        float dx = hv.x - cv.x, dy = hv.y - cv.y;
        float dz = hv.z - cv.z, dw = hv.w - cv.w;
        acc1 = fmaf(dx, dx, acc1);
        acc1 = fmaf(dy, dy, acc1);
        acc1 = fmaf(dz, dz, acc1);
        acc1 = fmaf(dw, dw, acc1);
    }

    // Pre-scale by 1/(count+1): both lanes and both partials of a row share inv.
    acc0 *= inv;
    acc1 *= inv;

    // WMMA reduction: A = ones(16x4), B = partials(4x16, N = lane%16), C = 0.
    v2f a; a.x = 1.0f; a.y = 1.0f;
    v2f b; b.x = acc0; b.y = acc1;
    v8f cacc = {};
    v8f d = __builtin_amdgcn_wmma_f32_16x16x4_f32(
        /*neg_a=*/false, a, /*neg_b=*/false, b,
        /*c_mod=*/(short)0, cacc, /*reuse_a=*/false, /*reuse_b=*/false);

    // D VGPR0: lanes 0-15 -> D[0][lane] = dist[lane]/(count[lane]+1);
    // lanes 16-31 mirror (D[8][lane-16], all D rows identical).
    float v = d[0];
#pragma unroll
    for (int off = 8; off > 0; off >>= 1)
        v += __shfl_down(v, off, 16);             // sum lanes 0..15 into lane 0

    if (lane == 0) atomicAdd(accum, v);
}

// ---------------- finalize: loss = 0.5 * s / (s + eps) -----------------------
__global__ void ccl_final_kernel(const float* __restrict__ accum,
                                 float* __restrict__ out) {
    float s = *accum;
    out[0] = 0.5f * s / (s + 1e-6f);
}

extern "C" void kernel_launch(void* const* d_in, const int* in_sizes, int n_in,
                              void* d_out, int out_size, void* d_ws, size_t ws_size,
                              hipStream_t stream) {
    const int*   y       = (const int*)d_in[0];
    const float* hidden  = (const float*)d_in[1];
    const float* centers = (const float*)d_in[2];
    float* out = (float*)d_out;

    unsigned int* hist  = (unsigned int*)d_ws;          // [0, NCLS)
    float*        accum = (float*)hist + NCLS;          // [NCLS]

    ccl_zero_kernel<<<1, 1024, 0, stream>>>(hist);
    ccl_hist_kernel<<<BATCH / 256, 256, 0, stream>>>(y, hist);

    // 65536 rows / 16 rows-per-wave = 4096 waves; 8 waves per 256-thread block.
    ccl_main_kernel<<<BATCH / 16 / 8, 256, 0, stream>>>(y, hidden, centers, hist, accum);

    ccl_final_kernel<<<1, 1, 0, stream>>>(accum, out);
}